// VanillaPLIF_13494787244788
// MI455X (gfx1250) — compile-verified
//
#include <hip/hip_runtime.h>

// PLIF forward scan, (T=32, B*N=2M) float32.
// Memory-bound: 512 MB total traffic -> ~22us floor at 23.3 TB/s.
// Strategy: one thread owns 4 neurons (float4), keeps membrane v in VGPRs
// across all T steps; B128 NT stores; gfx1250 async global->LDS
// double-buffering of the next timestep (ASYNCcnt-ordered, per-lane slots).

typedef __attribute__((ext_vector_type(4))) float f32x4;
typedef int v4i __attribute__((vector_size(16)));
typedef __attribute__((address_space(1))) v4i v4i_g;   // global
typedef __attribute__((address_space(3))) v4i v4i_l;   // LDS

#define T_STEPS 32
#define BLOCK   256

#if defined(__has_builtin)
# if __has_builtin(__builtin_amdgcn_global_load_async_to_lds_b128) && \
     __has_builtin(__builtin_amdgcn_s_wait_asynccnt)
#  define USE_ASYNC_LDS 1
# endif
#endif
#ifndef USE_ASYNC_LDS
# define USE_ASYNC_LDS 0
#endif

__device__ __forceinline__ f32x4 plif_step(f32x4& v, f32x4 xt, float beta) {
  // v = beta*v + x ; s = (v >= 1) ; v = s ? 0 : v   (hard reset)
  v = beta * v + xt;
  f32x4 s;
  s.x = (v.x >= 1.0f) ? 1.0f : 0.0f;
  s.y = (v.y >= 1.0f) ? 1.0f : 0.0f;
  s.z = (v.z >= 1.0f) ? 1.0f : 0.0f;
  s.w = (v.w >= 1.0f) ? 1.0f : 0.0f;
  v.x = (v.x >= 1.0f) ? 0.0f : v.x;
  v.y = (v.y >= 1.0f) ? 0.0f : v.y;
  v.z = (v.z >= 1.0f) ? 0.0f : v.z;
  v.w = (v.w >= 1.0f) ? 0.0f : v.w;
  return s;
}

__global__ __launch_bounds__(BLOCK) void plif_scan_vec(
    const f32x4* __restrict__ xin,      // (T, n4) of float4
    const float* __restrict__ beta_raw, // scalar
    f32x4* __restrict__ out,            // (T, n4) of float4
    int n4)
{
  const int i = blockIdx.x * BLOCK + threadIdx.x;
  if (i >= n4) return;

  const float beta = 1.0f / (1.0f + __expf(-beta_raw[0]));
  f32x4 v = {0.0f, 0.0f, 0.0f, 0.0f};

#if USE_ASYNC_LDS
  // Per-lane double buffer: lane L only ever touches stage[*][L], so only
  // ASYNCcnt ordering is needed (async ops complete in order) -- no barriers.
  __shared__ __align__(16) f32x4 stage[2][BLOCK];
  const unsigned tid = threadIdx.x;

  // Prologue: kick off t=0.
  __builtin_amdgcn_global_load_async_to_lds_b128(
      (v4i_g*)(xin + i), (v4i_l*)&stage[0][tid], 0, 0);

  for (int t = 0; t < T_STEPS - 1; ++t) {
    // Issue t+1, then wait until only it remains in flight -> buffer t ready.
    __builtin_amdgcn_global_load_async_to_lds_b128(
        (v4i_g*)(xin + (size_t)(t + 1) * n4 + i),
        (v4i_l*)&stage[(t + 1) & 1][tid], 0, 0);
    __builtin_amdgcn_s_wait_asynccnt(1);

    f32x4 xt = stage[t & 1][tid];        // ds_load_b128
    f32x4 s  = plif_step(v, xt, beta);
    __builtin_nontemporal_store(s, out + (size_t)t * n4 + i);
  }
  // Epilogue: drain the last buffer.
  __builtin_amdgcn_s_wait_asynccnt(0);
  {
    f32x4 xt = stage[(T_STEPS - 1) & 1][tid];
    f32x4 s  = plif_step(v, xt, beta);
    __builtin_nontemporal_store(s, out + (size_t)(T_STEPS - 1) * n4 + i);
  }
#else
  #pragma unroll 4
  for (int t = 0; t < T_STEPS; ++t) {
    if (t + 2 < T_STEPS)  // global_prefetch_b8 two timesteps ahead (8 MB stride)
      __builtin_prefetch((const void*)(xin + (size_t)(t + 2) * n4 + i), 0, 1);
    f32x4 xt = __builtin_nontemporal_load(xin + (size_t)t * n4 + i);
    f32x4 s  = plif_step(v, xt, beta);
    __builtin_nontemporal_store(s, out + (size_t)t * n4 + i);
  }
#endif
}

// Scalar tail for BN % 4 != 0 (never fires for the reference shape).
__global__ void plif_scan_tail(
    const float* __restrict__ x, const float* __restrict__ beta_raw,
    float* __restrict__ out, long long bn, long long start)
{
  long long i = start + (long long)blockIdx.x * blockDim.x + threadIdx.x;
  if (i >= bn) return;
  const float beta = 1.0f / (1.0f + __expf(-beta_raw[0]));
  float v = 0.0f;
  for (int t = 0; t < T_STEPS; ++t) {
    float xt = x[(size_t)t * bn + i];
    v = beta * v + xt;
    float s = (v >= 1.0f) ? 1.0f : 0.0f;
    out[(size_t)t * bn + i] = s;
    v = (v >= 1.0f) ? 0.0f : v;
  }
}

extern "C" void kernel_launch(void* const* d_in, const int* in_sizes, int n_in,
                              void* d_out, int out_size, void* d_ws, size_t ws_size,
                              hipStream_t stream) {
  (void)n_in; (void)out_size; (void)d_ws; (void)ws_size;

  const float* x    = (const float*)d_in[0];   // (T, B, N) fp32
  const float* braw = (const float*)d_in[1];   // scalar fp32
  float*       out  = (float*)d_out;           // (T, B, N) fp32

  const long long total = in_sizes[0];
  const long long bn    = total / T_STEPS;     // B*N
  const long long n4    = bn >> 2;             // float4 groups

  if (n4 > 0) {
    dim3 grid((unsigned)((n4 + BLOCK - 1) / BLOCK));
    plif_scan_vec<<<grid, BLOCK, 0, stream>>>(
        (const f32x4*)x, braw, (f32x4*)out, (int)n4);
  }
  const long long rem_start = n4 << 2;
  if (rem_start < bn) {
    long long rem = bn - rem_start;
    plif_scan_tail<<<dim3((unsigned)((rem + 255) / 256)), 256, 0, stream>>>(
        x, braw, out, bn, rem_start);
  }
}